// DeformableILM_35854386987265
// MI455X (gfx1250) — compile-verified
//
#include <hip/hip_runtime.h>
#include <math.h>

typedef __attribute__((ext_vector_type(2))) float v2f;
typedef __attribute__((ext_vector_type(8))) float v8f;

#define BB 2
#define CC 64
#define HH 128
#define WW 128
#define HW (HH * WW)     // 16384
#define MTOT (BB * HW)   // 32768

__device__ __forceinline__ float4 fma4(float a, const float4 v, const float4 c) {
  return make_float4(fmaf(a, v.x, c.x), fmaf(a, v.y, c.y),
                     fmaf(a, v.z, c.z), fmaf(a, v.w, c.w));
}

__device__ __forceinline__ float gelu_exact(float v) {
  return 0.5f * v * (1.0f + erff(v * 0.70710678118654752f));
}

// ---------------- x: NCHW -> NHWC ----------------
__global__ __launch_bounds__(256) void nchw_to_nhwc(const float* __restrict__ x,
                                                    float* __restrict__ xt) {
  int t = blockIdx.x * 256 + threadIdx.x;   // t = ((b*HW + hw)*64 + c), total 2^21
  int c  = t & 63;
  int hw = (t >> 6) & (HW - 1);
  int b  = t >> 20;
  xt[t] = x[(size_t)(b * CC + c) * HW + hw];
}

// ---------------- out = x (residual) ----------------
__global__ __launch_bounds__(256) void copy_res(const float4* __restrict__ x,
                                                float4* __restrict__ o) {
  int t = blockIdx.x * 256 + threadIdx.x;
  o[t] = x[t];
}

// ---------------- weight repack: (N,C,K,K) -> [kk][n][c], zero-pad n>=N2 ----------------
__global__ __launch_bounds__(256) void repack_kernel(const float* __restrict__ src,
                                                     float* __restrict__ dst,
                                                     int K2, int N2, int Npad) {
  int total = K2 * Npad * CC;
  int idx = blockIdx.x * 256 + threadIdx.x;
  if (idx >= total) return;
  int kk = idx / (Npad * CC);
  int r  = idx - kk * (Npad * CC);
  int n  = r / CC;
  int c  = r - n * CC;
  dst[idx] = (n < N2) ? src[(size_t)(n * CC + c) * K2 + kk] : 0.0f;
}

// ---------------- raw offsets -> absolute sample positions ----------------
__global__ __launch_bounds__(256) void positions_kernel(const float* __restrict__ offraw,
                                                        const float* __restrict__ boff,
                                                        float2* __restrict__ pos,
                                                        int K, int PAD, int N2) {
  int K2 = K * K;
  int total = BB * K2 * HW;
  int idx = blockIdx.x * 256 + threadIdx.x;
  if (idx >= total) return;
  int b    = idx / (K2 * HW);
  int rest = idx - b * (K2 * HW);
  int kk   = rest >> 14;          // / HW
  int hw   = rest & (HW - 1);
  int h = hw >> 7, w = hw & (WW - 1);
  int m = b * HW + hw;
  float oy = offraw[(size_t)m * N2 + 2 * kk]     + boff[2 * kk];
  float ox = offraw[(size_t)m * N2 + 2 * kk + 1] + boff[2 * kk + 1];
  int ky = kk / K, kx = kk - ky * K;
  pos[idx] = make_float2((float)(h + ky - PAD) + oy, (float)(w + kx - PAD) + ox);
}

// ---------------- implicit-GEMM conv via f32 WMMA ----------------
// MODE 0: im2col (offset conv), writes raw conv result to offraw[m][N2]
// MODE 1: deformable (bilinear sample via pos), out[nchw] += gelu(acc)
template <int K, int PAD, int NT, int MODE>
__global__ __launch_bounds__(256)
void deform_gemm(const float* __restrict__ xt,       // [B][H][W][C]
                 const float2* __restrict__ pos,     // MODE1: [B][K2][HW]
                 const float* __restrict__ wr,       // [K2][NT*16][64]
                 float* __restrict__ out,            // MODE1
                 float* __restrict__ offraw,         // MODE0
                 int N2) {
  constexpr int K2 = K * K;
  constexpr int NCOR = (MODE == 1) ? 4 : 1;
  __shared__ float lds_w[NT * 16 * 64];

  const int tid  = threadIdx.x;
  const int lane = tid & 31;
  const int wv   = tid >> 5;
  const int l16  = lane & 15;
  const bool hiHalf = (lane >= 16);

  const int m0 = blockIdx.x * 128 + wv * 16;  // 16 pixels per wave
  const int p  = m0 + l16;                    // this lane's pixel row of A
  const int b  = p >> 14;                     // / HW
  const int hw = p & (HW - 1);
  const int h  = hw >> 7;
  const int wx = hw & (WW - 1);
  const int cbB = hiHalf ? 8 : 0;             // channel sub-base (A/B k-slot permutation)

  v8f acc[NT];
#pragma unroll
  for (int t = 0; t < NT; ++t)
    acc[t] = (v8f){0.f, 0.f, 0.f, 0.f, 0.f, 0.f, 0.f, 0.f};

  const float4* xt4 = (const float4*)xt;

  for (int kk = 0; kk < K2; ++kk) {
    // cooperative LDS load of this tap's weight slab [n][c]
    {
      const float4* src = (const float4*)(wr + (size_t)kk * (NT * 16 * 64));
      float4* dst = (float4*)lds_w;
#pragma unroll
      for (int i = 0; i < NT; ++i) dst[tid + i * 256] = src[tid + i * 256];
    }
    __syncthreads();

    // per-lane sampling setup: corner bases + bilinear weights (weight=0 kills OOB)
    unsigned sbase[NCOR];
    float swgt[NCOR];
    if (MODE == 1) {
      float2 pq = pos[(size_t)(b * K2 + kk) * HW + hw];
      float py = pq.x, px = pq.y;
      float y0f = floorf(py), x0f = floorf(px);
      float ly = py - y0f, lx = px - x0f;
      int y0 = (int)y0f, x0 = (int)x0f;
      float wy[2]  = {1.0f - ly, ly};
      float wxf[2] = {1.0f - lx, lx};
#pragma unroll
      for (int cy = 0; cy < 2; ++cy) {
#pragma unroll
        for (int cx = 0; cx < 2; ++cx) {
          int yy = y0 + cy, xx = x0 + cx;
          bool ok = (yy >= 0) && (yy < HH) && (xx >= 0) && (xx < WW);
          int yc = yy < 0 ? 0 : (yy > HH - 1 ? HH - 1 : yy);
          int xc = xx < 0 ? 0 : (xx > WW - 1 ? WW - 1 : xx);
          sbase[cy * 2 + cx] = ((unsigned)(b * HH + yc) * WW + xc) << 6;
          swgt[cy * 2 + cx]  = ok ? (wy[cy] * wxf[cx]) : 0.0f;
        }
      }
    } else {
      int ky = kk / K, kx = kk - (kk / K) * K;
      int yy = h + ky - PAD, xx = wx + kx - PAD;
      bool ok = (yy >= 0) && (yy < HH) && (xx >= 0) && (xx < WW);
      int yc = yy < 0 ? 0 : (yy > HH - 1 ? HH - 1 : yy);
      int xc = xx < 0 ? 0 : (xx > WW - 1 ? WW - 1 : xx);
      sbase[0] = ((unsigned)(b * HH + yc) * WW + xc) << 6;
      swgt[0]  = ok ? 1.0f : 0.0f;
    }

    // 4 channel groups of 16; each group feeds 4 WMMAs per n-tile.
    // Channel permutation: lanes 0-15 carry channels cg*16+0..7 (k-slots 0,1),
    // lanes 16-31 carry cg*16+8..15 (k-slots 2,3); B loaded identically.
#pragma unroll
    for (int cg = 0; cg < 4; ++cg) {
      const int cb = cg * 16 + cbB;
      float4 lo  = {0.f, 0.f, 0.f, 0.f};
      float4 hi4 = {0.f, 0.f, 0.f, 0.f};
#pragma unroll
      for (int cr = 0; cr < NCOR; ++cr) {
        unsigned e = (sbase[cr] + cb) >> 2;   // float4 index
        float g = swgt[cr];
        lo  = fma4(g, xt4[e],     lo);
        hi4 = fma4(g, xt4[e + 1], hi4);
      }
      float a8[8] = {lo.x, lo.y, lo.z, lo.w, hi4.x, hi4.y, hi4.z, hi4.w};
#pragma unroll
      for (int j = 0; j < 4; ++j) {
        v2f A;
        A.x = a8[2 * j];
        A.y = a8[2 * j + 1];
#pragma unroll
        for (int t = 0; t < NT; ++t) {
          const v2f Bv =
              *(const v2f*)&lds_w[(t * 16 + l16) * 64 + cg * 16 + cbB + 2 * j];
          acc[t] = __builtin_amdgcn_wmma_f32_16x16x4_f32(
              false, A, false, Bv, (short)0, acc[t], false, false);
        }
      }
    }
    __syncthreads();
  }

  // epilogue: C/D layout -> VGPR r, lane L: M = r + 8*(L>=16), N = L%16
#pragma unroll
  for (int t = 0; t < NT; ++t) {
#pragma unroll
    for (int r = 0; r < 8; ++r) {
      int mrow = m0 + r + (hiHalf ? 8 : 0);
      int n = t * 16 + l16;
      float v = acc[t][r];
      if (MODE == 1) {
        int b2 = mrow >> 14, hw2 = mrow & (HW - 1);
        size_t idx = (size_t)(b2 * CC + n) * HW + hw2;
        out[idx] += gelu_exact(v);
      } else {
        if (n < N2) offraw[(size_t)mrow * N2 + n] = v;
      }
    }
  }
}

// ---------------- per-K pipeline ----------------
template <int K, int PAD, int NTO>
static void run_K(const float* xt, const float* woffw, const float* boffv,
                  const float* wmainw, float* out, float* posf, float* offraw,
                  float* wrm, float* wro, hipStream_t s) {
  constexpr int K2 = K * K, N2 = 2 * K2, Npad = NTO * 16;
  repack_kernel<<<(K2 * Npad * CC + 255) / 256, 256, 0, s>>>(woffw, wro, K2, N2, Npad);
  repack_kernel<<<(K2 * 64 * CC + 255) / 256, 256, 0, s>>>(wmainw, wrm, K2, 64, 64);
  deform_gemm<K, PAD, NTO, 0><<<MTOT / 128, 256, 0, s>>>(
      xt, nullptr, wro, nullptr, offraw, N2);
  positions_kernel<<<(BB * K2 * HW + 255) / 256, 256, 0, s>>>(
      offraw, boffv, (float2*)posf, K, PAD, N2);
  deform_gemm<K, PAD, 4, 1><<<MTOT / 128, 256, 0, s>>>(
      xt, (const float2*)posf, wrm, out, nullptr, 0);
}

extern "C" void kernel_launch(void* const* d_in, const int* in_sizes, int n_in,
                              void* d_out, int out_size, void* d_ws, size_t ws_size,
                              hipStream_t stream) {
  const float* x = (const float*)d_in[0];
  const float* woff[3]  = {(const float*)d_in[1], (const float*)d_in[4], (const float*)d_in[7]};
  const float* boff[3]  = {(const float*)d_in[2], (const float*)d_in[5], (const float*)d_in[8]};
  const float* wmain[3] = {(const float*)d_in[3], (const float*)d_in[6], (const float*)d_in[9]};
  float* out = (float*)d_out;
  float* ws  = (float*)d_ws;

  // workspace layout (floats)
  float* xt      = ws;                    // 2,097,152  (NHWC x, 8 MB)
  float* posf    = xt + 2097152;          // 3,211,264  (float2 positions, reused per K)
  float* offraw  = posf + 3211264;        // 3,211,264  (raw offset-conv out, reused)
  float* wrm     = offraw + 3211264;      //   200,704  (repacked main weights, max K=7)
  float* wro     = wrm + 200704;          //   351,232  (repacked offset weights, max K=7)
  // total ~9.07M floats = 36.3 MB

  nchw_to_nhwc<<<(BB * HW * CC) / 256, 256, 0, stream>>>(x, xt);
  copy_res<<<(BB * CC * HW / 4) / 256, 256, 0, stream>>>((const float4*)x, (float4*)out);

  run_K<3, 1, 2>(xt, woff[0], boff[0], wmain[0], out, posf, offraw, wrm, wro, stream);
  run_K<5, 2, 4>(xt, woff[1], boff[1], wmain[1], out, posf, offraw, wrm, wro, stream);
  run_K<7, 3, 7>(xt, woff[2], boff[2], wmain[2], out, posf, offraw, wrm, wro, stream);
}